// SkipGCN_15556371546755
// MI455X (gfx1250) — compile-verified
//
#include <hip/hip_runtime.h>
#include <cstdint>

typedef __attribute__((ext_vector_type(2))) float v2f;
typedef __attribute__((ext_vector_type(8))) float v8f;

#define D_IN 128   // feature dim (d == hidden == 128)
#define NCLS 2

// ---------------------------------------------------------------- degree ----
__global__ __launch_bounds__(256) void k_set1(float* __restrict__ p, int n) {
  int i = blockIdx.x * blockDim.x + threadIdx.x;
  if (i < n) p[i] = 1.0f;   // +1 self-loop baked in
}

__global__ __launch_bounds__(256) void k_deg_scatter(const long long* __restrict__ dst,
                                                     float* __restrict__ deg, int E) {
  int e = blockIdx.x * blockDim.x + threadIdx.x;
  if (e < E) atomicAdd(&deg[(int)dst[e]], 1.0f);
}

__global__ __launch_bounds__(256) void k_rsqrt(float* __restrict__ p, int n) {
  int i = blockIdx.x * blockDim.x + threadIdx.x;
  if (i < n) p[i] = 1.0f / sqrtf(p[i]);   // deg^-0.5, full-precision
}

// ------------------------------------------------- xh = x @ W1 via WMMA -----
// One wave computes one 16x16 tile of xh. A: 16x4 f32 fragment (v2f/lane),
// B: 4x16 f32 fragment (v2f/lane), C/D: 16x16 f32 (v8f/lane). K=128 -> 32 wmma.
__global__ __launch_bounds__(256) void k_gemm1_wmma(const float* __restrict__ x,
                                                    const float* __restrict__ W,
                                                    float* __restrict__ xh, int M) {
  const int warp = blockIdx.x * (blockDim.x >> 5) + (threadIdx.x >> 5);
  const int mt = warp >> 3;        // 8 N-tiles (128/16) per M-tile
  const int nt = warp & 7;
  const int mtiles = (M + 15) >> 4;
  if (mt >= mtiles) return;        // wave-uniform: EXEC stays all-ones

  const int lane    = threadIdx.x & 31;
  const int halfsel = lane >> 4;   // lanes 16-31 handle K+2,K+3 (A) / rows K+2,K+3 (B)
  const int lidx    = lane & 15;
  int rowA = mt * 16 + lidx;
  if (rowA >= M) rowA = M - 1;     // clamp (M=50000 is a multiple of 16; safety only)
  const int colB = nt * 16 + lidx;

  v8f acc = {};
  #pragma unroll
  for (int k0 = 0; k0 < D_IN; k0 += 4) {
    const int ka = k0 + (halfsel << 1);
    v2f a = *(const v2f*)(x + (size_t)rowA * D_IN + ka);        // 8B aligned
    v2f b;
    b.x = W[(size_t)ka * D_IN + colB];
    b.y = W[(size_t)(ka + 1) * D_IN + colB];
    acc = __builtin_amdgcn_wmma_f32_16x16x4_f32(
        /*neg_a=*/false, a, /*neg_b=*/false, b,
        /*c_mod=*/(short)0, acc, /*reuse_a=*/false, /*reuse_b=*/false);
  }

  // D layout: VGPR v -> row mt*16 + v + 8*halfsel, col = colB
  #pragma unroll
  for (int v = 0; v < 8; ++v) {
    const int r = mt * 16 + v + (halfsel << 3);
    if (r < M) xh[(size_t)r * D_IN + colB] = acc[v];
  }
}

// ----------------------------------- agg1 = xh * dinv^2 + b1 (self-loop) ----
__global__ __launch_bounds__(256) void k_agg1_init(const float* __restrict__ xh,
                                                   const float* __restrict__ dinv,
                                                   const float* __restrict__ b1,
                                                   float* __restrict__ agg, int N) {
  int idx = blockIdx.x * blockDim.x + threadIdx.x;
  if (idx >= N * D_IN) return;
  const int i = idx >> 7;
  const int k = idx & (D_IN - 1);
  const float di = dinv[i];
  agg[idx] = xh[idx] * (di * di) + b1[k];
}

// --------------------- conv1 scatter: agg1[dst] += xh[src]*norm (wave/edge) -
__global__ __launch_bounds__(256) void k_edge1(const long long* __restrict__ src,
                                               const long long* __restrict__ dst,
                                               const float* __restrict__ dinv,
                                               const float* __restrict__ xh,
                                               float* __restrict__ agg, int E) {
  const int e = blockIdx.x * (blockDim.x >> 5) + (threadIdx.x >> 5);
  if (e >= E) return;
  const int lane = threadIdx.x & 31;
  const int s = (int)src[e];
  const int d = (int)dst[e];
  const float norm = dinv[s] * dinv[d];
  const float4 v = *(const float4*)(xh + (size_t)s * D_IN + lane * 4);   // coalesced 512B/wave
  float* p = agg + (size_t)d * D_IN + lane * 4;
  atomicAdd(p + 0, v.x * norm);
  atomicAdd(p + 1, v.y * norm);
  atomicAdd(p + 2, v.z * norm);
  atomicAdd(p + 3, v.w * norm);
}

// ------ wave/node: g = [x, relu(h)] @ W2 ; out = g*dinv^2 + b2 + x@W_skip ---
__global__ __launch_bounds__(256) void k_node2(const float* __restrict__ x,
                                               const float* __restrict__ hraw,
                                               const float* __restrict__ dinv,
                                               const float* __restrict__ W2,
                                               const float* __restrict__ b2,
                                               const float* __restrict__ Wsk,
                                               float* __restrict__ g,
                                               float* __restrict__ out, int N) {
  const int i = blockIdx.x * (blockDim.x >> 5) + (threadIdx.x >> 5);
  if (i >= N) return;
  const int lane = threadIdx.x & 31;
  const float4 xv = *(const float4*)(x    + (size_t)i * D_IN + lane * 4);
  float4 hv       = *(const float4*)(hraw + (size_t)i * D_IN + lane * 4);
  hv.x = fmaxf(hv.x, 0.0f); hv.y = fmaxf(hv.y, 0.0f);
  hv.z = fmaxf(hv.z, 0.0f); hv.w = fmaxf(hv.w, 0.0f);

  float g0 = 0.f, g1 = 0.f, s0 = 0.f, s1 = 0.f;
  const float xs[4] = {xv.x, xv.y, xv.z, xv.w};
  const float hs[4] = {hv.x, hv.y, hv.z, hv.w};
  #pragma unroll
  for (int j = 0; j < 4; ++j) {
    const int k = lane * 4 + j;
    g0 += xs[j] * W2[k * NCLS + 0] + hs[j] * W2[(D_IN + k) * NCLS + 0];
    g1 += xs[j] * W2[k * NCLS + 1] + hs[j] * W2[(D_IN + k) * NCLS + 1];
    s0 += xs[j] * Wsk[k * NCLS + 0];
    s1 += xs[j] * Wsk[k * NCLS + 1];
  }
  #pragma unroll
  for (int off = 16; off > 0; off >>= 1) {   // wave32 butterfly reduction
    g0 += __shfl_xor(g0, off, 32);
    g1 += __shfl_xor(g1, off, 32);
    s0 += __shfl_xor(s0, off, 32);
    s1 += __shfl_xor(s1, off, 32);
  }
  if (lane == 0) {
    const float di = dinv[i];
    const float dd = di * di;
    g[i * NCLS + 0] = g0;
    g[i * NCLS + 1] = g1;
    out[i * NCLS + 0] = g0 * dd + b2[0] + s0;
    out[i * NCLS + 1] = g1 * dd + b2[1] + s1;
  }
}

// ---------------------- conv2 scatter: out[dst] += g[src]*norm (2 floats) ---
__global__ __launch_bounds__(256) void k_edge2(const long long* __restrict__ src,
                                               const long long* __restrict__ dst,
                                               const float* __restrict__ dinv,
                                               const float* __restrict__ g,
                                               float* __restrict__ out, int E) {
  const int e = blockIdx.x * blockDim.x + threadIdx.x;
  if (e >= E) return;
  const int s = (int)src[e];
  const int d = (int)dst[e];
  const float norm = dinv[s] * dinv[d];
  atomicAdd(&out[d * NCLS + 0], g[s * NCLS + 0] * norm);
  atomicAdd(&out[d * NCLS + 1], g[s * NCLS + 1] * norm);
}

// ---------------------------------------------------------------------------
extern "C" void kernel_launch(void* const* d_in, const int* in_sizes, int n_in,
                              void* d_out, int out_size, void* d_ws, size_t ws_size,
                              hipStream_t stream) {
  const float*     x   = (const float*)d_in[0];
  const long long* ei  = (const long long*)d_in[1];   // int64 [2, E]
  const float*     W1  = (const float*)d_in[2];
  const float*     b1  = (const float*)d_in[3];
  const float*     W2  = (const float*)d_in[4];
  const float*     b2  = (const float*)d_in[5];
  const float*     Wsk = (const float*)d_in[6];
  float*           out = (float*)d_out;

  const int N = in_sizes[0] / D_IN;   // 50000
  const int E = in_sizes[1] / 2;      // 1600000
  const long long* src = ei;
  const long long* dst = ei + E;

  // Workspace layout: deg/dinv | xh | agg1 | g  (~52 MB)
  float* deg = (float*)d_ws;
  float* xh  = deg + N;
  float* agg = xh + (size_t)N * D_IN;
  float* g   = agg + (size_t)N * D_IN;

  // 1) symmetric-norm degree
  k_set1<<<(N + 255) / 256, 256, 0, stream>>>(deg, N);
  k_deg_scatter<<<(E + 255) / 256, 256, 0, stream>>>(dst, deg, E);
  k_rsqrt<<<(N + 255) / 256, 256, 0, stream>>>(deg, N);   // deg -> dinv in place

  // 2) xh = x @ W1  (WMMA f32 16x16x4, 8 waves/block = one M-tile per block)
  const int mtiles = (N + 15) / 16;   // 3125
  k_gemm1_wmma<<<mtiles, 256, 0, stream>>>(x, W1, xh, N);

  // 3) conv1 self-loop + bias init, then edge scatter-add
  k_agg1_init<<<((N * D_IN) + 255) / 256, 256, 0, stream>>>(xh, deg, b1, agg, N);
  k_edge1<<<(E + 7) / 8, 256, 0, stream>>>(src, dst, deg, xh, agg, E);

  // 4) fused relu + h2@W2 + residual + bias + self-loop -> out init; g to ws
  k_node2<<<(N + 7) / 8, 256, 0, stream>>>(x, agg, deg, W2, b2, Wsk, g, out, N);

  // 5) conv2 edge scatter into out
  k_edge2<<<(E + 255) / 256, 256, 0, stream>>>(src, dst, deg, g, out, E);
}